// GCNTorch_64441689309902
// MI455X (gfx1250) — compile-verified
//
#include <hip/hip_runtime.h>
#include <math.h>

// ---------------------------------------------------------------------------
// 3-layer GCN for MI455X (gfx1250, wave32).
//   shared:    deg/dinv + CSR-by-dst (counts -> scan -> fill), built once.
//   per layer: h = x @ W            (WMMA f32 16x16x4, LDS-tiled)
//              out[i] = relu( b + sum_{e: dst=i} h[src]*norm + h[i]*dinv[i]^2 )
//              (bias + ReLU fused into the CSR gather; zero output atomics)
// ---------------------------------------------------------------------------

typedef float v2f __attribute__((ext_vector_type(2)));
typedef float v8f __attribute__((ext_vector_type(8)));

// ---------------- degree / normalization ----------------------------------
__global__ void deg_init_kernel(float* deg, int n) {
    int i = blockIdx.x * blockDim.x + threadIdx.x;
    if (i < n) deg[i] = 1.0f;  // self-loop weight
}

__global__ void deg_accum_kernel(const int* __restrict__ dst,
                                 const float* __restrict__ ew,
                                 float* deg, int E) {
    int e = blockIdx.x * blockDim.x + threadIdx.x;
    if (e < E) atomicAdd(&deg[dst[e]], ew[e]);
}

__global__ void dinv_kernel(float* deg, int n) {
    int i = blockIdx.x * blockDim.x + threadIdx.x;
    if (i < n) {
        float d = deg[i];
        deg[i] = (d > 0.0f) ? rsqrtf(d) : 0.0f;  // in-place deg -> dinv
    }
}

// ---------------- CSR build (by destination) -------------------------------
__global__ void cnt_init_kernel(int* cnt, int n) {
    int i = blockIdx.x * blockDim.x + threadIdx.x;
    if (i < n) cnt[i] = 0;
}

__global__ void cnt_accum_kernel(const int* __restrict__ dst, int* cnt, int E) {
    int e = blockIdx.x * blockDim.x + threadIdx.x;
    if (e < E) atomicAdd(&cnt[dst[e]], 1);
}

// single-block exclusive scan (100K elems, one-time cost, reused by 3 layers)
__global__ __launch_bounds__(1024)
void scan_kernel(const int* __restrict__ cnt, int* __restrict__ off, int n) {
    __shared__ int tmp[1024];
    __shared__ int carry_s;
    const int t = threadIdx.x;
    if (t == 0) carry_s = 0;
    __syncthreads();
    for (int base = 0; base < n; base += 1024) {
        int i = base + t;
        int v = (i < n) ? cnt[i] : 0;
        tmp[t] = v;
        __syncthreads();
        for (int s = 1; s < 1024; s <<= 1) {   // inclusive Hillis-Steele
            int add = (t >= s) ? tmp[t - s] : 0;
            __syncthreads();
            tmp[t] += add;
            __syncthreads();
        }
        if (i < n) off[i] = carry_s + tmp[t] - v;  // exclusive
        __syncthreads();
        if (t == 1023) carry_s += tmp[1023];
        __syncthreads();
    }
    if (t == 0) off[n] = carry_s;
}

__global__ void cursor_copy_kernel(const int* __restrict__ off, int* cur, int n) {
    int i = blockIdx.x * blockDim.x + threadIdx.x;
    if (i < n) cur[i] = off[i];
}

// scatter edges into CSR slots; norm precomputed so the hot loop is pure gather
__global__ void csr_fill_kernel(const int* __restrict__ src,
                                const int* __restrict__ dst,
                                const float* __restrict__ ew,
                                const float* __restrict__ dinv,
                                int* cur, int* __restrict__ csr_src,
                                float* __restrict__ csr_w, int E) {
    int e = blockIdx.x * blockDim.x + threadIdx.x;
    if (e < E) {
        int s = src[e], d = dst[e];
        int pos = atomicAdd(&cur[d], 1);
        csr_src[pos] = s;
        csr_w[pos]   = dinv[s] * ew[e] * dinv[d];
    }
}

// ---------------- WMMA f32 GEMM:  C[M,N] = A[M,K] * B[K,N] ----------------
// Block: 320 threads = 10 waves; block tile 80x32; wave tile 16x16.
// Requires M%80==0, N%32==0, K%16==0 (M=100000, N in {256,32}, K in {512,256}).
#define SA_STRIDE  18   // 80x16 A tile, padded: conflict-free ds_load_b64 frags
#define SBT_STRIDE 18   // B tile TRANSPOSED (32 cols x 16 K): b64 frags, no conflicts

__global__ __launch_bounds__(320)
void gemm_wmma_f32_kernel(const float* __restrict__ A,
                          const float* __restrict__ B,
                          float* __restrict__ C,
                          int M, int N, int K) {
    __shared__ float sA [80 * SA_STRIDE];
    __shared__ float sBT[32 * SBT_STRIDE];

    const int tid  = threadIdx.x;
    const int wave = tid >> 5;
    const int lane = tid & 31;
    const int mi   = wave % 5;      // 5 M-subtiles
    const int ni   = wave / 5;      // 2 N-subtiles
    const int half = lane >> 4;     // fragment K split (ISA 7.12.2)
    const int l15  = lane & 15;
    const int m0   = blockIdx.y * 80;
    const int n0   = blockIdx.x * 32;

    v8f acc = {};

    for (int k0 = 0; k0 < K; k0 += 16) {
        // --- stage A tile: 80x16 = 320 float4, one per thread -------------
        {
            int r  = tid >> 2;
            int c4 = (tid & 3) << 2;
            const float4 v = *(const float4*)(A + (size_t)(m0 + r) * K + k0 + c4);
            float* sp = &sA[r * SA_STRIDE + c4];          // 8B-aligned rows
            ((float2*)sp)[0] = make_float2(v.x, v.y);
            ((float2*)sp)[1] = make_float2(v.z, v.w);
        }
        // --- stage B tile transposed: 16x32 -> sBT[col][k] ----------------
        if (tid < 128) {
            int r  = tid >> 3;           // K row 0..15
            int c4 = (tid & 7) << 2;     // col 0,4,...,28
            const float4 v = *(const float4*)(B + (size_t)(k0 + r) * N + n0 + c4);
            sBT[(c4 + 0) * SBT_STRIDE + r] = v.x;
            sBT[(c4 + 1) * SBT_STRIDE + r] = v.y;
            sBT[(c4 + 2) * SBT_STRIDE + r] = v.z;
            sBT[(c4 + 3) * SBT_STRIDE + r] = v.w;
        }
        __syncthreads();

        // --- 4x V_WMMA_F32_16X16X4_F32 (K-step 16) ------------------------
#pragma unroll
        for (int kk = 0; kk < 4; ++kk) {
            const int ks = kk * 4 + half * 2;  // lanes 0-15: K {0,1}; 16-31: {2,3}
            v2f a = *(const v2f*)&sA [(mi * 16 + l15) * SA_STRIDE  + ks];
            v2f b = *(const v2f*)&sBT[(ni * 16 + l15) * SBT_STRIDE + ks];
            acc = __builtin_amdgcn_wmma_f32_16x16x4_f32(
                false, a, false, b, (short)0, acc, false, false);
        }
        __syncthreads();
    }

    // --- store C: VGPR v -> row v (lanes 0-15) / row v+8 (lanes 16-31) ----
    const int col   = n0 + ni * 16 + l15;
    const int rbase = m0 + mi * 16 + half * 8;
#pragma unroll
    for (int v = 0; v < 8; ++v) {
        C[(size_t)(rbase + v) * N + col] = acc[v];
    }
}

// ---------------- CSR aggregation: one wave32 per dst node ----------------
// out[node] = (relu)( bias + h[node]*dinv^2 + sum_edges h[src]*norm )
// Register accumulation, coalesced edge-metadata loads + __shfl broadcast.
template <int NF, bool RELU>
__global__ __launch_bounds__(256)
void aggregate_csr_kernel(const float* __restrict__ h,
                          float* __restrict__ out,
                          const int* __restrict__ off,
                          const int* __restrict__ csr_src,
                          const float* __restrict__ csr_w,
                          const float* __restrict__ dinv,
                          const float* __restrict__ bias,
                          int n) {
    const int node = (blockIdx.x * blockDim.x + threadIdx.x) >> 5;
    const int lane = threadIdx.x & 31;
    if (node >= n) return;

    constexpr int CH = (NF + 127) / 128;   // float4 chunks per lane
    const int f0 = lane * 4;

    float4 acc[CH];
#pragma unroll
    for (int c = 0; c < CH; ++c) {
        int f = f0 + c * 128;
        if (f < NF) acc[c] = *(const float4*)(bias + f);
    }

    // self-loop term
    {
        float di = dinv[node];
        float sw = di * di;
        const float* hp = h + (size_t)node * NF;
#pragma unroll
        for (int c = 0; c < CH; ++c) {
            int f = f0 + c * 128;
            if (f < NF) {
                float4 v = *(const float4*)(hp + f);
                acc[c].x += v.x * sw; acc[c].y += v.y * sw;
                acc[c].z += v.z * sw; acc[c].w += v.w * sw;
            }
        }
    }

    // incoming edges: batch 32 at a time (coalesced) then broadcast via shfl
    const int beg = off[node], end = off[node + 1];
    for (int p0 = beg; p0 < end; p0 += 32) {
        const int np = min(32, end - p0);
        int   sL = (p0 + lane < end) ? csr_src[p0 + lane] : 0;
        float wL = (p0 + lane < end) ? csr_w[p0 + lane]   : 0.0f;
        for (int j = 0; j < np; ++j) {
            int   s  = __shfl(sL, j);
            float wv = __shfl(wL, j);
            const float* sp = h + (size_t)s * NF;
#pragma unroll
            for (int c = 0; c < CH; ++c) {
                int f = f0 + c * 128;
                if (f < NF) {
                    float4 v = *(const float4*)(sp + f);
                    acc[c].x += v.x * wv; acc[c].y += v.y * wv;
                    acc[c].z += v.z * wv; acc[c].w += v.w * wv;
                }
            }
        }
    }

    float* op = out + (size_t)node * NF;
#pragma unroll
    for (int c = 0; c < CH; ++c) {
        int f = f0 + c * 128;
        if (f < NF) {
            if (RELU) {
                acc[c].x = fmaxf(acc[c].x, 0.0f); acc[c].y = fmaxf(acc[c].y, 0.0f);
                acc[c].z = fmaxf(acc[c].z, 0.0f); acc[c].w = fmaxf(acc[c].w, 0.0f);
            }
            *(float4*)(op + f) = acc[c];
        }
    }
}

// ---------------------------------------------------------------------------
extern "C" void kernel_launch(void* const* d_in, const int* in_sizes, int n_in,
                              void* d_out, int out_size, void* d_ws, size_t ws_size,
                              hipStream_t stream) {
    const float* features = (const float*)d_in[0];
    const int*   ei       = (const int*)d_in[1];   // int32 per harness contract
    const float* ew       = (const float*)d_in[2];
    const float* W0 = (const float*)d_in[3];
    const float* b0 = (const float*)d_in[4];
    const float* W1 = (const float*)d_in[5];
    const float* b1 = (const float*)d_in[6];
    const float* W2 = (const float*)d_in[7];
    const float* b2 = (const float*)d_in[8];

    const int inF = in_sizes[3] / in_sizes[4];   // 512
    const int hid = in_sizes[4];                 // 256
    const int cls = in_sizes[8];                 // 32
    const int Nn  = in_sizes[0] / inF;           // 100000
    const int E   = in_sizes[1] / 2;             // 3.2M
    const int* src = ei;
    const int* dst = ei + E;

    // ---- workspace carve-up (all 256B aligned) ----
    char* ws = (char*)d_ws;
    size_t o = 0;
    auto carve = [&](size_t bytes) {
        void* p = ws + o;
        o += (bytes + 255) & ~(size_t)255;
        return p;
    };
    float* dinv    = (float*)carve((size_t)Nn * 4);
    int*   cnt     = (int*)  carve((size_t)Nn * 4);
    int*   off     = (int*)  carve(((size_t)Nn + 1) * 4);
    int*   cur     = (int*)  carve((size_t)Nn * 4);
    int*   csr_src = (int*)  carve((size_t)E * 4);
    float* csr_w   = (float*)carve((size_t)E * 4);
    float* buf0    = (float*)carve((size_t)Nn * hid * 4);  // GEMM outputs
    float* buf1    = (float*)carve((size_t)Nn * hid * 4);  // activations
    float* out     = (float*)d_out;

    const int T = 256;
    const int nodeBlocks = (Nn + (T / 32) - 1) / (T / 32);   // 1 wave per node

    // ---- shared normalization + CSR (identical across all 3 layers) ----
    deg_init_kernel   <<<(Nn + T - 1) / T, T, 0, stream>>>(dinv, Nn);
    deg_accum_kernel  <<<(E  + T - 1) / T, T, 0, stream>>>(dst, ew, dinv, E);
    dinv_kernel       <<<(Nn + T - 1) / T, T, 0, stream>>>(dinv, Nn);
    cnt_init_kernel   <<<(Nn + T - 1) / T, T, 0, stream>>>(cnt, Nn);
    cnt_accum_kernel  <<<(E  + T - 1) / T, T, 0, stream>>>(dst, cnt, E);
    scan_kernel       <<<1, 1024, 0, stream>>>(cnt, off, Nn);
    cursor_copy_kernel<<<(Nn + T - 1) / T, T, 0, stream>>>(off, cur, Nn);
    csr_fill_kernel   <<<(E  + T - 1) / T, T, 0, stream>>>(src, dst, ew, dinv,
                                                           cur, csr_src, csr_w, E);

    // ---- layer 0: [Nn,512] x [512,256] ----
    gemm_wmma_f32_kernel<<<dim3(hid / 32, Nn / 80), 320, 0, stream>>>(
        features, W0, buf0, Nn, hid, inF);
    aggregate_csr_kernel<256, true><<<nodeBlocks, T, 0, stream>>>(
        buf0, buf1, off, csr_src, csr_w, dinv, b0, Nn);

    // ---- layer 1: [Nn,256] x [256,256] ----
    gemm_wmma_f32_kernel<<<dim3(hid / 32, Nn / 80), 320, 0, stream>>>(
        buf1, W1, buf0, Nn, hid, hid);
    aggregate_csr_kernel<256, true><<<nodeBlocks, T, 0, stream>>>(
        buf0, buf1, off, csr_src, csr_w, dinv, b1, Nn);

    // ---- layer 2: [Nn,256] x [256,32] -> d_out ----
    gemm_wmma_f32_kernel<<<dim3(cls / 32, Nn / 80), 320, 0, stream>>>(
        buf1, W2, buf0, Nn, cls, hid);
    aggregate_csr_kernel<32, false><<<nodeBlocks, T, 0, stream>>>(
        buf0, out, off, csr_src, csr_w, dinv, b2, Nn);
}